// BandedJointEncoder_27728308863076
// MI455X (gfx1250) — compile-verified
//
#include <hip/hip_runtime.h>
#include <hip/hip_bf16.h>
#include <math.h>

typedef float v2f __attribute__((ext_vector_type(2)));
typedef float v4f __attribute__((ext_vector_type(4)));
typedef float v8f __attribute__((ext_vector_type(8)));

#define BB 64
#define TT 256
#define DD 36
#define HH 128
#define OO 48
#define ZZ 16
#define KIM 108           // 3*36 im2col depth
#define TILE_ELEMS (TT*TT)   // 65536 floats per (b,z) tril tile

__device__ __forceinline__ float softplus_f(float x) {
    return (x > 20.f) ? x : log1pf(__expf(x));
}

// ---------------------------------------------------------------------------
// Kernel A: fused conv1d(k=3) + ReLU(GEMM 128x128) + GEMM 128x48 via
// V_WMMA_F32_16X16X4_F32. One block = one batch b, one 16-row tile of T.
// 8 waves; each wave owns one 16-column slab of H for conv/GEMM1; waves 0..2
// own the 48 output columns for GEMM2. Mean written straight to d_out;
// softplus channels stashed into the head of each (b,z) output tile.
// ---------------------------------------------------------------------------
__global__ __launch_bounds__(256) void encA(
    const float* __restrict__ x,  const float* __restrict__ cw,
    const float* __restrict__ cb, const float* __restrict__ w1,
    const float* __restrict__ b1, const float* __restrict__ w2,
    const float* __restrict__ b2,
    float* __restrict__ outMean, float* __restrict__ tril)
{
    __shared__ float Aim[16 * KIM];   // im2col tile (16 x 108)
    __shared__ float h1[16 * HH];     // conv output
    __shared__ float h2[16 * HH];     // relu(gemm1) output

    const int b   = blockIdx.y;
    const int t0  = blockIdx.x * 16;
    const int tid = threadIdx.x;

    // im2col row m == contiguous 108 floats starting at x[b, t0+m-1, 0]
    for (int e = tid; e < 16 * KIM; e += 256) {
        int m   = e / KIM;
        int k   = e - m * KIM;
        int off = (t0 - 1 + m) * DD + k;     // offset inside batch b
        float v = 0.f;
        if (off >= 0 && off < TT * DD) v = x[b * TT * DD + off];
        Aim[m * KIM + k] = v;
    }
    __syncthreads();

    const int w    = tid >> 5;
    const int lane = tid & 31;
    const int half = lane >> 4;     // 0: K pair {0,1}, 1: K pair {2,3}
    const int l16  = lane & 15;     // M (A) / N (B,C,D)
    const int kb   = half * 2;

    // ---- conv as GEMM: (16 x 108) x (108 x 128) ----
    {
        const int n = w * 16 + l16;
        v8f acc;
        float bias = cb[n];
        #pragma unroll
        for (int v = 0; v < 8; ++v) acc[v] = bias;
        #pragma unroll
        for (int ks = 0; ks < 27; ++ks) {
            int k = ks * 4 + kb;
            v2f a, bf;
            a[0]  = Aim[l16 * KIM + k];
            a[1]  = Aim[l16 * KIM + k + 1];
            bf[0] = cw[k * HH + n];
            bf[1] = cw[(k + 1) * HH + n];
            acc = __builtin_amdgcn_wmma_f32_16x16x4_f32(
                false, a, false, bf, (short)0, acc, false, false);
        }
        #pragma unroll
        for (int v = 0; v < 8; ++v) h1[(v + 8 * half) * HH + n] = acc[v];
    }
    __syncthreads();

    // ---- GEMM1: (16 x 128) x (128 x 128) + ReLU ----
    {
        const int n = w * 16 + l16;
        v8f acc;
        float bias = b1[n];
        #pragma unroll
        for (int v = 0; v < 8; ++v) acc[v] = bias;
        #pragma unroll
        for (int ks = 0; ks < 32; ++ks) {
            int k = ks * 4 + kb;
            v2f a, bf;
            a[0]  = h1[l16 * HH + k];
            a[1]  = h1[l16 * HH + k + 1];
            bf[0] = w1[k * HH + n];
            bf[1] = w1[(k + 1) * HH + n];
            acc = __builtin_amdgcn_wmma_f32_16x16x4_f32(
                false, a, false, bf, (short)0, acc, false, false);
        }
        #pragma unroll
        for (int v = 0; v < 8; ++v)
            h2[(v + 8 * half) * HH + n] = fmaxf(acc[v], 0.f);
    }
    __syncthreads();

    // ---- GEMM2: (16 x 128) x (128 x 48), waves 0..2, fused epilogue ----
    if (w < 3) {
        const int n = w * 16 + l16;
        v8f acc;
        float bias = b2[n];
        #pragma unroll
        for (int v = 0; v < 8; ++v) acc[v] = bias;
        #pragma unroll
        for (int ks = 0; ks < 32; ++ks) {
            int k = ks * 4 + kb;
            v2f a, bf;
            a[0]  = h2[l16 * HH + k];
            a[1]  = h2[l16 * HH + k + 1];
            bf[0] = w2[k * OO + n];
            bf[1] = w2[(k + 1) * OO + n];
            acc = __builtin_amdgcn_wmma_f32_16x16x4_f32(
                false, a, false, bf, (short)0, acc, false, false);
        }
        #pragma unroll
        for (int v = 0; v < 8; ++v) {
            int t = t0 + v + 8 * half;
            float val = acc[v];
            if (w == 0) {
                // mean[b, z, t] = mapped[b, t, z], z = l16
                outMean[((size_t)b * ZZ + l16) * TT + t] = val;
            } else {
                // channels 16..47 -> softplus, stash in own tril tile head
                int c = n - ZZ;            // 0..31
                int z = c >> 1, parity = c & 1;
                tril[((size_t)(b * ZZ + z)) * TILE_ELEMS + parity * TT + t] =
                    softplus_f(val);
            }
        }
    }
}

// ---------------------------------------------------------------------------
// Kernel B: per (b,z), closed-form inverse of upper-bidiagonal U.
//   scale_tril[i,j] = (-1)^(i-j) * exp(S[i]-S[j]) / d[i]   for j <= i, else 0
// with S = prefix sum of log(e_k/d_k). 256 threads, NT float4 stores,
// fully coalesced 1KB per row.
// ---------------------------------------------------------------------------
__global__ __launch_bounds__(256) void encB(float* __restrict__ tril)
{
    const int b   = blockIdx.y;
    const int z   = blockIdx.x;
    const int tid = threadIdx.x;
    float* tile = tril + (size_t)(b * ZZ + z) * TILE_ELEMS;

    // stash (written by encA): [0..255]=softplus(ch 2z), [256..511]=softplus(ch 2z+1)
    float dv = tile[tid] + 1.0f;         // diag d_t
    float ev = tile[TT + tid];           // super-diag e_t (t < 255 used)

    __shared__ float sc[256];
    __shared__ float S[256];
    __shared__ float invd[256];

    sc[tid] = (tid < TT - 1) ? __logf(ev / dv) : 0.0f;
    __syncthreads();
    #pragma unroll
    for (int off = 1; off < 256; off <<= 1) {   // inclusive Hillis-Steele scan
        float v = sc[tid];
        if (tid >= off) v += sc[tid - off];
        __syncthreads();
        sc[tid] = v;
        __syncthreads();
    }
    S[tid]    = (tid == 0) ? 0.f : sc[tid - 1]; // S[i] = sum_{k<i} log(e_k/d_k)
    invd[tid] = 1.0f / dv;
    __syncthreads();

    const int rsub = tid >> 6;        // 4 rows in flight per iteration
    const int j0   = (tid & 63) * 4;  // this thread's fixed 4 columns
    const float Sj0 = S[j0], Sj1 = S[j0 + 1], Sj2 = S[j0 + 2], Sj3 = S[j0 + 3];

    for (int it = 0; it < 64; ++it) {
        int i = it * 4 + rsub;
        float Si  = S[i];
        float inv = invd[i];
        v4f r;
        #pragma unroll
        for (int c = 0; c < 4; ++c) {
            int j = j0 + c;
            float Sj = (c == 0) ? Sj0 : (c == 1) ? Sj1 : (c == 2) ? Sj2 : Sj3;
            float v = 0.f;
            if (j <= i) {
                v = __expf(Si - Sj) * inv;
                if ((i - j) & 1) v = -v;
                if (!(fabsf(v) < 3.4e38f)) v = 0.f;   // kill inf/NaN like ref
            }
            r[c] = v;
        }
        __builtin_nontemporal_store(r, (v4f*)(tile + (size_t)i * TT + j0));
    }
}

extern "C" void kernel_launch(void* const* d_in, const int* in_sizes, int n_in,
                              void* d_out, int out_size, void* d_ws, size_t ws_size,
                              hipStream_t stream) {
    const float* x  = (const float*)d_in[0];
    const float* cw = (const float*)d_in[1];
    const float* cb = (const float*)d_in[2];
    const float* w1 = (const float*)d_in[3];
    const float* b1 = (const float*)d_in[4];
    const float* w2 = (const float*)d_in[5];
    const float* b2 = (const float*)d_in[6];

    float* outMean = (float*)d_out;                       // (64,16,256)
    float* tril    = outMean + (size_t)BB * ZZ * TT;      // (64,16,256,256)

    encA<<<dim3(TT / 16, BB), 256, 0, stream>>>(x, cw, cb, w1, b1, w2, b2,
                                                outMean, tril);
    encB<<<dim3(ZZ, BB), 256, 0, stream>>>(tril);
}